// MeanStdStiffRegularizer_43104291782998
// MI455X (gfx1250) — compile-verified
//
#include <hip/hip_runtime.h>
#include <hip/hip_bf16.h>
#include <math.h>

#define NSEG      512
#define EPS       1e-6f
#define STRENGTH  0.01f

typedef __attribute__((ext_vector_type(4))) float f4v;
typedef __attribute__((ext_vector_type(4))) int   i4v;
typedef __attribute__((ext_vector_type(2))) float v2f;
typedef __attribute__((ext_vector_type(8))) float v8f;

// ---------------------------------------------------------------------------
// Kernel 0: zero the 8KB scratch accumulators (3*512 f32 sums + 512 u32 counts)
// ---------------------------------------------------------------------------
__global__ __launch_bounds__(256) void zero_ws_kernel(float* __restrict__ ws) {
    int i = blockIdx.x * blockDim.x + threadIdx.x;
    if (i < 4 * NSEG) ws[i] = 0.0f;   // zero bit-pattern valid for u32 counts too
}

// ---------------------------------------------------------------------------
// Kernel 1: stream 128MB once; per-WG LDS segmented accumulation, then merge
// into global accumulators with native f32 atomics.
// ---------------------------------------------------------------------------
__global__ __launch_bounds__(256) void seg_accum_kernel(
    const f4v* __restrict__ x4, const i4v* __restrict__ id4,
    float* __restrict__ gsum, float* __restrict__ glog,
    float* __restrict__ glog2, unsigned* __restrict__ gcnt, int n4) {

    __shared__ float    s_sum [NSEG];
    __shared__ float    s_log [NSEG];
    __shared__ float    s_log2[NSEG];
    __shared__ unsigned s_cnt [NSEG];

    for (int i = threadIdx.x; i < NSEG; i += blockDim.x) {
        s_sum[i] = 0.0f; s_log[i] = 0.0f; s_log2[i] = 0.0f; s_cnt[i] = 0u;
    }
    __syncthreads();

    const int stride = gridDim.x * blockDim.x;
    for (int i = blockIdx.x * blockDim.x + threadIdx.x; i < n4; i += stride) {
        // Single-use stream: non-temporal 128-bit loads, prefetch next tile.
        f4v xv = __builtin_nontemporal_load(&x4[i]);
        i4v iv = __builtin_nontemporal_load(&id4[i]);
        if (i + stride < n4) {
            __builtin_prefetch(&x4[i + stride], 0, 0);
            __builtin_prefetch(&id4[i + stride], 0, 0);
        }
#pragma unroll
        for (int j = 0; j < 4; ++j) {
            float x = xv[j];
            int   s = iv[j];
            float l = __logf(fabsf(x) + EPS);
            atomicAdd(&s_sum [s], x);        // ds_add_f32
            atomicAdd(&s_log [s], l);        // ds_add_f32
            atomicAdd(&s_log2[s], l * l);    // ds_add_f32
            atomicAdd(&s_cnt [s], 1u);       // ds_add_u32
        }
    }
    __syncthreads();

    for (int i = threadIdx.x; i < NSEG; i += blockDim.x) {
        unsafeAtomicAdd(&gsum [i], s_sum [i]);   // global_atomic_add_f32
        unsafeAtomicAdd(&glog [i], s_log [i]);
        unsafeAtomicAdd(&glog2[i], s_log2[i]);
        atomicAdd(&gcnt[i], s_cnt[i]);           // global_atomic_add_u32
    }
}

// ---------------------------------------------------------------------------
// Kernel 2: single wave32. Compute 512 per-segment loss terms and sum them
// with V_WMMA_F32_16X16X4_F32: A = ones(16x4), B = terms(4x16).
// D[i][j] = sum_k B[k][j]  -> a pure sum, invariant to B's slot layout.
// 8 accumulating WMMAs cover 8*64 = 512 terms; reduce D row 0 across lanes.
// ---------------------------------------------------------------------------
__device__ __forceinline__ float loss_term(
    int s, const float* gsum, const float* glog, const float* glog2,
    const unsigned* gcnt, const float* tm, const float* ts) {
    float cnt  = fmaxf((float)gcnt[s], 1.0f);
    float inv  = 1.0f / cnt;
    float mean = gsum[s]  * inv;
    float ml   = glog[s]  * inv;
    float var  = fmaxf(glog2[s] * inv - ml * ml, 0.0f);   // one-pass variance
    float stdv = sqrtf(var + EPS);
    float dm   = mean - tm[s];
    float dsd  = stdv - ts[s];
    return 0.5f * (dm * dm + dsd * dsd);   // (1-STD_WEIGHT), STD_WEIGHT = 0.5
}

__global__ __launch_bounds__(32) void finalize_kernel(
    const float* __restrict__ gsum, const float* __restrict__ glog,
    const float* __restrict__ glog2, const unsigned* __restrict__ gcnt,
    const float* __restrict__ tm, const float* __restrict__ ts,
    float* __restrict__ out) {

    const int lane = threadIdx.x;   // one full wave32, EXEC all ones

    v2f a; a.x = 1.0f; a.y = 1.0f;  // A = ones(16x4): every slot = 1.0
    v8f c = {};
#pragma unroll
    for (int it = 0; it < 8; ++it) {
        int s0 = it * 64 + lane;    // this wave's 64 terms per iteration
        v2f b;
        b.x = loss_term(s0,      gsum, glog, glog2, gcnt, tm, ts);
        b.y = loss_term(s0 + 32, gsum, glog, glog2, gcnt, tm, ts);
        // 8 args: (neg_a, A, neg_b, B, c_mod, C, reuse_a, reuse_b)
        c = __builtin_amdgcn_wmma_f32_16x16x4_f32(
            false, a, false, b, (short)0, c, false, false);
    }

    // D row M=0 lives in C VGPR0, lanes 0..15 (N = lane). Sum the 16 columns.
    float v = c[0];
#pragma unroll
    for (int off = 8; off >= 1; off >>= 1) v += __shfl_xor(v, off, 32);

    if (lane == 0) out[0] = v * (STRENGTH / (float)NSEG);
}

// ---------------------------------------------------------------------------
extern "C" void kernel_launch(void* const* d_in, const int* in_sizes, int n_in,
                              void* d_out, int out_size, void* d_ws, size_t ws_size,
                              hipStream_t stream) {
    const float* x   = (const float*)d_in[0];
    const int*   idx = (const int*)  d_in[1];
    const float* tm  = (const float*)d_in[2];
    const float* ts  = (const float*)d_in[3];

    float*    ws    = (float*)d_ws;
    float*    gsum  = ws;
    float*    glog  = ws + NSEG;
    float*    glog2 = ws + 2 * NSEG;
    unsigned* gcnt  = (unsigned*)(ws + 3 * NSEG);

    const int n  = in_sizes[0];
    const int n4 = n / 4;    // N_EDGES = 16777216, divisible by 4

    zero_ws_kernel<<<(4 * NSEG + 255) / 256, 256, 0, stream>>>(ws);

    // 2048 blocks * 256 threads * float4 = 8 grid-stride iterations over 16M.
    seg_accum_kernel<<<2048, 256, 0, stream>>>(
        (const f4v*)x, (const i4v*)idx, gsum, glog, glog2, gcnt, n4);

    finalize_kernel<<<1, 32, 0, stream>>>(
        gsum, glog, glog2, gcnt, tm, ts, (float*)d_out);
}